// LocalMLPBlock_86655260164510
// MI455X (gfx1250) — compile-verified
//
#include <hip/hip_runtime.h>
#include <hip/hip_bf16.h>

typedef __attribute__((ext_vector_type(2))) float v2f;
typedef __attribute__((ext_vector_type(8))) float v8f;

#define B_    8
#define S_    4096
#define K_    32
#define CIN   64
#define C1    67          // Cin + 3
#define MID   64
#define COUT  128
#define ROWS  (B_*S_*K_)  // 1048576
#define NWAVE (ROWS/16)   // 65536
#define BLK   256
#define WPB   (BLK/32)
#define GRID  (NWAVE/WPB) // 8192
#define W1STR 68          // padded row stride (even -> 8B-aligned b64 LDS loads)

__device__ __forceinline__ v8f wmma4(v2f a, v2f b, v8f c) {
  // D(16x16) = A(16x4) * B(4x16) + C, all f32
  return __builtin_amdgcn_wmma_f32_16x16x4_f32(false, a, false, b, (short)0, c, false, false);
}

// gemm1 (transposed): acc[t] tile holds y1[m][row], m = t*16 + 8*hi + v, row = lane&15.
// A = W1 tile (from LDS), B = x^T where x = [neigh_feat | rel_xyz | 0pad].
// All 16 global B-fragments are preloaded so the backend can keep many loads in
// flight (staggered s_wait_loadcnt) under the WMMA stream.
__device__ __forceinline__ void gemm1_frags(
    const float* __restrict__ nfeat, const float* __restrict__ nxyz,
    const float* __restrict__ cxyz, const float* __restrict__ w1s,
    int rowbase, int lane, v8f acc[4])
{
  const int n  = lane & 15;
  const int hi = lane >> 4;
  const int r  = rowbase + n;   // this lane's bsk row (N slot)
  const int bs = r >> 5;        // b*S + s
  const float* xrow = nfeat + (size_t)r * CIN;

  v2f xv[16];
#pragma unroll
  for (int ks = 0; ks < 16; ++ks)
    xv[ks] = *(const v2f*)(xrow + ks*4 + 2*hi);   // x[row][k0+2hi .. +1]

  const float rel0 = nxyz[(size_t)r*3+0] - cxyz[(size_t)bs*3+0];
  const float rel1 = nxyz[(size_t)r*3+1] - cxyz[(size_t)bs*3+1];
  const float rel2 = nxyz[(size_t)r*3+2] - cxyz[(size_t)bs*3+2];

  acc[0] = {}; acc[1] = {}; acc[2] = {}; acc[3] = {};
#pragma unroll
  for (int ks = 0; ks < 17; ++ks) {
    const int k0 = ks * 4;
    v2f xb;
    if (ks < 16) {
      xb = xv[ks];
    } else {                                      // c = 64..67 : rel + zero pad
      xb.x = hi ? rel2 : rel0;
      xb.y = hi ? 0.0f : rel1;
    }
#pragma unroll
    for (int t = 0; t < 4; ++t) {
      v2f wa = *(const v2f*)(w1s + (t*16 + n)*W1STR + k0 + 2*hi);
      acc[t] = wmma4(wa, xb, acc[t]);
    }
  }
}

// gemm2: D2 = y * W2^T (row-major result): acc2[t2] element (lane l, reg v) =
//   y2[row = v + 8*hi][o = t2*16 + (l&15)].
// The y A-fragment is rebuilt from gemm1 D-fragments via a uniform half-broadcast
// shuffle (same gather serves as A here because A/B WMMA layouts are transposes).
__device__ __forceinline__ void gemm2_frags(const float* __restrict__ w2s,
                                            int lane, const v8f y[4], v8f acc2[8])
{
  const int n  = lane & 15;
  const int hi = lane >> 4;
#pragma unroll
  for (int t2 = 0; t2 < 8; ++t2) acc2[t2] = {};
#pragma unroll
  for (int ks = 0; ks < 16; ++ks) {
    const int k0 = ks * 4;
    const int t  = k0 >> 4;         // source tile
    const int hn = (k0 >> 3) & 1;   // half that owns channels k0..k0+3
    const int va = k0 & 7;          // base reg at source
    const int src = n + (hn << 4);
    // row n's channels k0..k0+3 gathered from the owning half
    const float c0 = __shfl(y[t][va+0], src, 32);
    const float c1 = __shfl(y[t][va+1], src, 32);
    const float c2 = __shfl(y[t][va+2], src, 32);
    const float c3 = __shfl(y[t][va+3], src, 32);
    v2f xb;                          // A-frag: y[row=n][k0+2hi .. +1]
    xb.x = hi ? c2 : c0;
    xb.y = hi ? c3 : c1;
#pragma unroll
    for (int t2 = 0; t2 < 8; ++t2) {
      v2f wb = *(const v2f*)(w2s + (t2*16 + n)*MID + k0 + 2*hi);  // B-frag: W2^T
      acc2[t2] = wmma4(xb, wb, acc2[t2]);
    }
  }
}

__device__ __forceinline__ void fill_w1(float* w1s, const float* __restrict__ W1) {
  for (int i = threadIdx.x; i < 64*W1STR; i += BLK) {
    int m = i / W1STR, c = i % W1STR;
    w1s[i] = (c < C1) ? W1[m*C1 + c] : 0.0f;
  }
}

// ---------------- phase 1: gemm1 -> GroupNorm1 partial stats ----------------
__global__ void __launch_bounds__(BLK) k_gemm1_stats(
    const float* __restrict__ cxyz, const float* __restrict__ nxyz,
    const float* __restrict__ nfeat, const float* __restrict__ W1,
    float* __restrict__ sum1, float* __restrict__ sumsq1)
{
  __shared__ float w1s[64*W1STR];
  fill_w1(w1s, W1);
  __syncthreads();
  const int lane = threadIdx.x & 31;
  const int wid  = blockIdx.x * WPB + (threadIdx.x >> 5);
  const int rowbase = wid * 16;
  const int b = rowbase >> 17;          // / (S*K)
  const int n = lane & 15, hi = lane >> 4;
  v8f acc[4];
  gemm1_frags(nfeat, nxyz, cxyz, w1s, rowbase, lane, acc);
#pragma unroll
  for (int t = 0; t < 4; ++t) {          // tile t covers groups 2t (hi=0), 2t+1 (hi=1)
    float s = 0.f, ss = 0.f;
#pragma unroll
    for (int v = 0; v < 8; ++v) { float x = acc[t][v]; s += x; ss += x*x; }
#pragma unroll
    for (int m = 1; m <= 8; m <<= 1) { s += __shfl_xor(s, m, 32); ss += __shfl_xor(ss, m, 32); }
    if (n == 0) {
      const int g = 2*t + hi;
      atomicAdd(&sum1[b*8 + g], s);
      atomicAdd(&sumsq1[b*8 + g], ss);
    }
  }
}

// ---------------- fold stats -> per-(b,channel) scale/shift ----------------
__global__ void k_finalize(const float* __restrict__ sum, const float* __restrict__ sumsq,
                           const float* __restrict__ gam, const float* __restrict__ bet,
                           float* __restrict__ scale, float* __restrict__ shift,
                           int C, float invN)
{
  const int idx = threadIdx.x;          // 64 = (b,g)
  if (idx >= 64) return;
  const int b = idx >> 3, g = idx & 7, cg = C >> 3;
  const float mean = sum[idx] * invN;
  const float var  = sumsq[idx] * invN - mean*mean;
  const float rstd = rsqrtf(var + 1e-5f);
  for (int j = 0; j < cg; ++j) {
    const int c = g*cg + j;
    const float sc = rstd * gam[c];
    scale[b*C + c] = sc;
    shift[b*C + c] = bet[c] - mean * sc;
  }
}

// ---------------- phase 2: gemm1 -> norm1+relu -> gemm2 -> stats2 ----------------
__global__ void __launch_bounds__(BLK) k_gemm2_stats(
    const float* __restrict__ cxyz, const float* __restrict__ nxyz,
    const float* __restrict__ nfeat, const float* __restrict__ W1,
    const float* __restrict__ W2,
    const float* __restrict__ scale1, const float* __restrict__ shift1,
    float* __restrict__ sum2, float* __restrict__ sumsq2)
{
  __shared__ float w1s[64*W1STR];
  __shared__ float w2s[COUT*MID];
  fill_w1(w1s, W1);
  for (int i = threadIdx.x; i < COUT*MID; i += BLK) w2s[i] = W2[i];
  __syncthreads();
  const int lane = threadIdx.x & 31;
  const int wid  = blockIdx.x * WPB + (threadIdx.x >> 5);
  const int rowbase = wid * 16;
  const int b = rowbase >> 17;
  const int hi = lane >> 4;
  v8f acc[4];
  gemm1_frags(nfeat, nxyz, cxyz, w1s, rowbase, lane, acc);
  const float* sc1 = scale1 + b*MID + 8*hi;
  const float* sh1 = shift1 + b*MID + 8*hi;
#pragma unroll
  for (int t = 0; t < 4; ++t)
#pragma unroll
    for (int v = 0; v < 8; ++v) {
      const int m = t*16 + v;
      acc[t][v] = fmaxf(0.0f, acc[t][v]*sc1[m] + sh1[m]);
    }
  v8f acc2[8];
  gemm2_frags(w2s, lane, acc, acc2);
#pragma unroll
  for (int t2 = 0; t2 < 8; ++t2) {       // GN2 group == tile index (16 ch/group)
    float s = 0.f, ss = 0.f;
#pragma unroll
    for (int v = 0; v < 8; ++v) { float x = acc2[t2][v]; s += x; ss += x*x; }
#pragma unroll
    for (int m = 1; m <= 16; m <<= 1) { s += __shfl_xor(s, m, 32); ss += __shfl_xor(ss, m, 32); }
    if (lane == 0) {
      atomicAdd(&sum2[b*8 + t2], s);
      atomicAdd(&sumsq2[b*8 + t2], ss);
    }
  }
}

// ---------------- phase 3: full pipeline -> norm2+relu -> max over K ----------------
__global__ void __launch_bounds__(BLK) k_final(
    const float* __restrict__ cxyz, const float* __restrict__ nxyz,
    const float* __restrict__ nfeat, const float* __restrict__ W1,
    const float* __restrict__ W2,
    const float* __restrict__ scale1, const float* __restrict__ shift1,
    const float* __restrict__ scale2, const float* __restrict__ shift2,
    unsigned* __restrict__ out)
{
  __shared__ float w1s[64*W1STR];
  __shared__ float w2s[COUT*MID];
  fill_w1(w1s, W1);
  for (int i = threadIdx.x; i < COUT*MID; i += BLK) w2s[i] = W2[i];
  __syncthreads();
  const int lane = threadIdx.x & 31;
  const int wid  = blockIdx.x * WPB + (threadIdx.x >> 5);
  const int rowbase = wid * 16;
  const int b  = rowbase >> 17;
  const int bs = rowbase >> 5;           // b*S + s (wave = one K-half of one (b,s))
  const int n = lane & 15, hi = lane >> 4;
  v8f acc[4];
  gemm1_frags(nfeat, nxyz, cxyz, w1s, rowbase, lane, acc);
  const float* sc1 = scale1 + b*MID + 8*hi;
  const float* sh1 = shift1 + b*MID + 8*hi;
#pragma unroll
  for (int t = 0; t < 4; ++t)
#pragma unroll
    for (int v = 0; v < 8; ++v) {
      const int m = t*16 + v;
      acc[t][v] = fmaxf(0.0f, acc[t][v]*sc1[m] + sh1[m]);
    }
  v8f acc2[8];
  gemm2_frags(w2s, lane, acc, acc2);
  // acc2[t2] element (lane, v) = y2[row = v + 8*hi][o = t2*16 + n]:
  // rows live in regs + hi, so the K-maxpool is 7 in-reg fmax + one shfl_xor(16).
#pragma unroll
  for (int t2 = 0; t2 < 8; ++t2) {
    const float sc = scale2[b*COUT + t2*16 + n];
    const float sh = shift2[b*COUT + t2*16 + n];
    float x = 0.0f;                      // relu output >= 0 -> 0 is the max identity
#pragma unroll
    for (int v = 0; v < 8; ++v)
      x = fmaxf(x, fmaxf(0.0f, acc2[t2][v]*sc + sh));
    x = fmaxf(x, __shfl_xor(x, 16, 32)); // combine rows 0-7 with rows 8-15
    if (hi == 0) {
      // f32 bits monotonic as u32 for non-negative values; out pre-zeroed
      atomicMax(out + (size_t)bs*COUT + (t2*16 + n), __float_as_uint(x));
    }
  }
}

extern "C" void kernel_launch(void* const* d_in, const int* in_sizes, int n_in,
                              void* d_out, int out_size, void* d_ws, size_t ws_size,
                              hipStream_t stream)
{
  const float* cxyz  = (const float*)d_in[0];
  // d_in[1] = center_feat: unused by the reference
  const float* nxyz  = (const float*)d_in[2];
  const float* nfeat = (const float*)d_in[3];
  const float* W1    = (const float*)d_in[4];
  const float* g1    = (const float*)d_in[5];
  const float* b1    = (const float*)d_in[6];
  const float* W2    = (const float*)d_in[7];
  const float* g2    = (const float*)d_in[8];
  const float* b2    = (const float*)d_in[9];

  float* ws     = (float*)d_ws;
  float* sum1   = ws + 0;      // 64
  float* sumsq1 = ws + 64;     // 64
  float* sum2   = ws + 128;    // 64
  float* sumsq2 = ws + 192;    // 64
  float* scale1 = ws + 256;    // 8*64
  float* shift1 = ws + 768;    // 8*64
  float* scale2 = ws + 1280;   // 8*128
  float* shift2 = ws + 2304;   // 8*128 (ends at 3328 floats ~13KB)

  hipMemsetAsync(d_ws, 0, 256 * sizeof(float), stream);               // stats accumulators
  hipMemsetAsync(d_out, 0, (size_t)out_size * sizeof(float), stream); // maxpool identity

  k_gemm1_stats<<<GRID, BLK, 0, stream>>>(cxyz, nxyz, nfeat, W1, sum1, sumsq1);
  k_finalize<<<1, 64, 0, stream>>>(sum1, sumsq1, g1, b1, scale1, shift1,
                                   MID, 1.0f / (float)(S_*K_*8));
  k_gemm2_stats<<<GRID, BLK, 0, stream>>>(cxyz, nxyz, nfeat, W1, W2,
                                          scale1, shift1, sum2, sumsq2);
  k_finalize<<<1, 64, 0, stream>>>(sum2, sumsq2, g2, b2, scale2, shift2,
                                   COUT, 1.0f / (float)(S_*K_*16));
  k_final<<<GRID, BLK, 0, stream>>>(cxyz, nxyz, nfeat, W1, W2,
                                    scale1, shift1, scale2, shift2,
                                    (unsigned*)d_out);
}